// QuaildGraphCut_7997229105730
// MI455X (gfx1250) — compile-verified
//
#include <hip/hip_runtime.h>

typedef __attribute__((ext_vector_type(2))) float v2f;
typedef __attribute__((ext_vector_type(8))) float v8f;

#define LAMBD 0.5f
#define N_ROWS 16384
#define N_COLS 512
#define COL_TILES 32                      // 512 / 16
#define ROW_GROUPS 64
#define ROWS_PER_WAVE (N_ROWS / ROW_GROUPS)   // 256
#define WAVES_PER_BLOCK 8
#define BLOCK_THREADS 256
#define TOTAL_WAVES (COL_TILES * ROW_GROUPS)  // 2048
#define NUM_BLOCKS (TOTAL_WAVES / WAVES_PER_BLOCK) // 256

// Stage 1: each wave owns a (16-column tile, 256-row group) slab of both a and b.
// Column sums are computed as ones(16x4) x B(4x16) accumulated with
// V_WMMA_F32_16X16X4_F32. D[m][n] = sum_k B[k][n] (replicated over m), so
// VGPR0 of the accumulator holds the 16 partial column sums, n = lane%16.
__global__ void __launch_bounds__(BLOCK_THREADS)
colsum_wmma_kernel(const float* __restrict__ a,
                   const float* __restrict__ b,
                   float* __restrict__ pa,
                   float* __restrict__ pb) {
  const int wave  = threadIdx.x >> 5;      // wave32
  const int lane  = threadIdx.x & 31;
  const int gwave = blockIdx.x * WAVES_PER_BLOCK + wave;

  const int col_tile  = gwave & (COL_TILES - 1);
  const int row_group = gwave >> 5;        // gwave / COL_TILES
  const int col0 = col_tile * 16;
  const int row0 = row_group * ROWS_PER_WAVE;

  const int n     = lane & 15;             // column within tile (B/D: N = lane%16)
  const int rhalf = lane >> 4;             // row half within the 4-row B tile

  const float* ap = a + (size_t)(row0 + rhalf) * N_COLS + col0 + n;
  const float* bp = b + (size_t)(row0 + rhalf) * N_COLS + col0 + n;

  v8f acc_a = {};
  v8f acc_b = {};
  v2f ones  = {1.0f, 1.0f};

  for (int r = 0; r < ROWS_PER_WAVE; r += 4) {
    v2f Ba, Bb;
    Ba.x = ap[0];                  // rows row0+r+rhalf
    Ba.y = ap[2 * N_COLS];         // rows row0+r+2+rhalf
    Bb.x = bp[0];
    Bb.y = bp[2 * N_COLS];
    // (neg_a, A, neg_b, B, c_mod, C, reuse_a, reuse_b)
    acc_a = __builtin_amdgcn_wmma_f32_16x16x4_f32(
        false, ones, false, Ba, (short)0, acc_a, false, false);
    acc_b = __builtin_amdgcn_wmma_f32_16x16x4_f32(
        false, ones, false, Bb, (short)0, acc_b, false, false);
    ap += 4 * N_COLS;
    bp += 4 * N_COLS;
  }

  // Every lane's VGPR0 holds colsum[col0 + lane%16]; lanes 0-15 write.
  if (lane < 16) {
    pa[row_group * N_COLS + col0 + n] = acc_a[0];
    pb[row_group * N_COLS + col0 + n] = acc_b[0];
  }
}

// Stage 2: deterministic cross-group reduction + dot product. One block.
__global__ void __launch_bounds__(512)
final_dot_kernel(const float* __restrict__ pa,
                 const float* __restrict__ pb,
                 float* __restrict__ out) {
  __shared__ float red[512];
  const int c = threadIdx.x;               // one thread per column
  float sa = 0.0f, sb = 0.0f;
  for (int g = 0; g < ROW_GROUPS; ++g) {
    sa += pa[g * N_COLS + c];
    sb += pb[g * N_COLS + c];
  }
  red[c] = sa * sb;
  __syncthreads();
  for (int s = 256; s > 0; s >>= 1) {
    if (c < s) red[c] += red[c + s];
    __syncthreads();
  }
  if (c == 0) out[0] = 2.0f * LAMBD * red[0];
}

extern "C" void kernel_launch(void* const* d_in, const int* in_sizes, int n_in,
                              void* d_out, int out_size, void* d_ws, size_t ws_size,
                              hipStream_t stream) {
  const float* a = (const float*)d_in[0];
  const float* b = (const float*)d_in[1];
  float* pa = (float*)d_ws;                      // [ROW_GROUPS][N_COLS]
  float* pb = pa + ROW_GROUPS * N_COLS;          // [ROW_GROUPS][N_COLS]

  colsum_wmma_kernel<<<NUM_BLOCKS, BLOCK_THREADS, 0, stream>>>(a, b, pa, pb);
  final_dot_kernel<<<1, 512, 0, stream>>>(pa, pb, (float*)d_out);
}